// PairwiseEmbeddings_17325898072624
// MI455X (gfx1250) — compile-verified
//
#include <hip/hip_runtime.h>

typedef float v2f __attribute__((ext_vector_type(2)));
typedef float v8f __attribute__((ext_vector_type(8)));

#define B_  2
#define L_  256
#define H_  256
#define A_  32
#define P_  64
#define NJP (L_ * P_)          // 16384 columns of the big GEMM
#define BDY 8

// ---------------------------------------------------------------------------
// Kernel 1: h[b,l,a] = sum_h emb[b,l,h] * in_w[a,h] + in_b[a]   (16384 outputs)
// ---------------------------------------------------------------------------
__global__ __launch_bounds__(256) void in_proj_kernel(
    const float* __restrict__ emb, const float* __restrict__ in_w,
    const float* __restrict__ in_b, float* __restrict__ h) {
  int idx = blockIdx.x * blockDim.x + threadIdx.x;
  if (idx >= B_ * L_ * A_) return;
  int a = idx & (A_ - 1);
  int l = (idx >> 5) & (L_ - 1);
  int b = idx >> 13;
  const float* e = emb + (b * L_ + l) * H_;
  const float* w = in_w + a * H_;
  float acc = in_b[a];
#pragma unroll 8
  for (int k = 0; k < H_; ++k) acc += e[k] * w[k];
  h[idx] = acc;
}

// ---------------------------------------------------------------------------
// Kernel 2: T[b,a,j*P+p] = sum_c out_w[p, a*A+c] * h[b,j,c]   (1M outputs)
// ---------------------------------------------------------------------------
__global__ __launch_bounds__(256) void mid_proj_kernel(
    const float* __restrict__ h, const float* __restrict__ out_w,
    float* __restrict__ T) {
  int idx = blockIdx.x * blockDim.x + threadIdx.x;
  if (idx >= B_ * A_ * NJP) return;
  int p = idx & (P_ - 1);
  int j = (idx >> 6) & (L_ - 1);
  int a = (idx >> 14) & (A_ - 1);
  int b = idx >> 19;
  const float* hr = h + (b * L_ + j) * A_;
  const float* wr = out_w + p * (A_ * A_) + a * A_;
  float acc = 0.f;
#pragma unroll
  for (int c = 0; c < A_; ++c) acc += hr[c] * wr[c];
  T[idx] = acc;  // idx == ((b*A_+a)*NJP) + j*P_ + p
}

// ---------------------------------------------------------------------------
// Kernel 3 (WMMA): out[b,i,j,p] = sum_a h[b,i,a] * T[b,a,j*P+p]
//                                 + out_b[p] + pos_b[p] + pos_w[p, clip(i-j)+8]
// Per-wave 16x16 tile of (i x jp); K=32 via 8 steps of V_WMMA_F32_16X16X4_F32.
// ---------------------------------------------------------------------------
__global__ __launch_bounds__(256) void pairwise_wmma_kernel(
    const float* __restrict__ h, const float* __restrict__ T,
    const float* __restrict__ out_b, const float* __restrict__ pos_w,
    const float* __restrict__ pos_b, float* __restrict__ out) {
  const int TILES_N = NJP / 16;  // 1024
  int wid  = (blockIdx.x * blockDim.x + threadIdx.x) >> 5;
  int lane = threadIdx.x & 31;
  int tn = wid & (TILES_N - 1);
  int tm = (wid >> 10) & 15;
  int b  = wid >> 14;
  if (b >= B_) return;

  const int n0 = tn * 16;
  const int i0 = tm * 16;
  const int lo = lane & 15;
  const int hi = lane >> 4;  // 0 or 1

  // A fragment source: row = i0+lo, K pair base offset 2*hi within each k-step
  const float* arow = h + (b * L_ + (i0 + lo)) * A_;
  // B fragment source: column = n0+lo, rows k-step*4 + 2*hi (+0,+1)
  const float* Tb = T + b * (A_ * NJP) + (n0 + lo);

  v8f acc = {};
#pragma unroll
  for (int kk = 0; kk < A_ / 4; ++kk) {
    const int kbase = kk * 4 + 2 * hi;
    v2f afrag = *(const v2f*)(arow + kbase);
    v2f bfrag;
    bfrag[0] = Tb[(kbase + 0) * NJP];
    bfrag[1] = Tb[(kbase + 1) * NJP];
    // (neg_a, A, neg_b, B, c_mod, C, reuse_a, reuse_b)
    acc = __builtin_amdgcn_wmma_f32_16x16x4_f32(
        false, afrag, false, bfrag, (short)0, acc, false, false);
  }

  // Epilogue: tile spans 16 consecutive jp within one j (P=64, n0 % 16 == 0)
  const int j = n0 >> 6;              // constant over tile
  const int p = (n0 & (P_ - 1)) + lo; // 16 consecutive p per half-wave
  const float bias = out_b[p] + pos_b[p];
  const float* pw = pos_w + p * (2 * BDY + 1);
  float* orow = out + ((size_t)(b * L_) * L_) * P_ + (size_t)j * P_ + p;

#pragma unroll
  for (int r = 0; r < 8; ++r) {
    int i = i0 + r + 8 * hi;
    int d = i - j;
    d = d < -BDY ? -BDY : (d > BDY ? BDY : d);
    orow[(size_t)i * (L_ * P_)] = acc[r] + bias + pw[d + BDY];
  }
}

// ---------------------------------------------------------------------------
extern "C" void kernel_launch(void* const* d_in, const int* in_sizes, int n_in,
                              void* d_out, int out_size, void* d_ws, size_t ws_size,
                              hipStream_t stream) {
  const float* emb   = (const float*)d_in[0];  // [2,256,256]
  const float* in_w  = (const float*)d_in[1];  // [32,256]
  const float* in_b  = (const float*)d_in[2];  // [32]
  const float* out_w = (const float*)d_in[3];  // [64,1024]
  const float* out_b = (const float*)d_in[4];  // [64]
  const float* pos_w = (const float*)d_in[5];  // [64,17]
  const float* pos_b = (const float*)d_in[6];  // [64]
  float* out = (float*)d_out;                  // [2,256,256,64]

  float* h = (float*)d_ws;                     // 16384 floats
  float* T = h + (B_ * L_ * A_);               // 1048576 floats (4 MB)

  in_proj_kernel<<<(B_ * L_ * A_ + 255) / 256, 256, 0, stream>>>(emb, in_w, in_b, h);
  mid_proj_kernel<<<(B_ * A_ * NJP + 255) / 256, 256, 0, stream>>>(h, out_w, T);

  // 2 * 16 * 1024 = 32768 waves, 8 waves per 256-thread block -> 4096 blocks
  pairwise_wmma_kernel<<<4096, 256, 0, stream>>>(h, T, out_b, pos_w, pos_b, out);
}